// Model_87497073754830
// MI455X (gfx1250) — compile-verified
//
#include <hip/hip_runtime.h>
#include <math.h>

typedef __attribute__((ext_vector_type(2))) float v2f;
typedef __attribute__((ext_vector_type(8))) float v8f;

#define N_ROWS 2000000
#define N_INF  128
#define N_HIDD 32
#define CT_SZ  16384
#define NCT    16
#define NBAGS  1024

// order-preserving float<->uint encoding for atomicMax on floats
__device__ __forceinline__ unsigned encf(float f) {
    int i = __float_as_int(f);
    return (unsigned)(i ^ ((i >> 31) | 0x80000000));
}
__device__ __forceinline__ float decf(unsigned u) {
    int i = ((int)u < 0) ? (int)(u ^ 0x80000000u) : (int)(~u);
    return __int_as_float(i);
}

// ---------------- K0: init accumulators ----------------
__global__ void k_init(unsigned* __restrict__ segMax, float* __restrict__ segDen,
                       float* __restrict__ P) {
    int i = blockIdx.x * 256 + threadIdx.x;
    if (i < CT_SZ) { segMax[i] = 0u; segDen[i] = 0.0f; }
    if (i < CT_SZ * N_HIDD) P[i] = 0.0f;
}

// ---------------- K1: H = relu(X@W1+b1) via f32 WMMA; s = H@wc+bc; seg max ----------------
__global__ __launch_bounds__(128) void k_gemm(
        const float* __restrict__ X, const float* __restrict__ W1,
        const float* __restrict__ b1, const float* __restrict__ wc,
        const float* __restrict__ bc, const int* __restrict__ batch,
        float* __restrict__ H, float* __restrict__ S, unsigned* __restrict__ segMax)
{
    // W1 stored TRANSPOSED: sWt[col*132 + row] so a lane's B pair (row kb, kb+1)
    // at fixed col is one contiguous 8-byte LDS load -> feeds WMMA with no movs.
    __shared__ float sWt[N_HIDD * 132];
    __shared__ float sX[4][16 * 132];     // per-wave 16x128 X tile, row stride 132

    const int tid  = threadIdx.x;
    const int wv   = tid >> 5;            // wave in block (0..3)
    const int lane = tid & 31;
    const int l16  = lane & 15;
    const int koff = (lane < 16) ? 0 : 2; // half-wave K offset per WMMA f32 layout

    // stage W1 (128x32) transposed
    for (int i = tid; i < N_INF * N_HIDD; i += 128) {
        int row = i >> 5, col = i & 31;
        sWt[col * 132 + row] = W1[i];
    }

    // stage this wave's 16x128 X tile (coalesced float4 loads)
    const long rowBase = (long)blockIdx.x * 64 + (long)wv * 16;
    const float4* Xv = (const float4*)X;
    for (int r = 0; r < 16; ++r) {
        float4 v = Xv[(rowBase + r) * 32 + lane];
        float* dst = &sX[wv][r * 132 + lane * 4];
        dst[0] = v.x; dst[1] = v.y; dst[2] = v.z; dst[3] = v.w;
    }
    __syncthreads();

    v8f acc0 = {};  // cols 0..15
    v8f acc1 = {};  // cols 16..31
    const float* xrow  = &sX[wv][l16 * 132];
    const float* wrow0 = &sWt[l16 * 132];          // B tile 0 (cols 0..15)
    const float* wrow1 = &sWt[(16 + l16) * 132];   // B tile 1 (cols 16..31)

    for (int k = 0; k < 32; ++k) {
        const int kb = 4 * k + koff;               // 8B-aligned offsets
        v2f a   = *(const v2f*)&xrow[kb];
        v2f b0  = *(const v2f*)&wrow0[kb];
        v2f b1v = *(const v2f*)&wrow1[kb];
        acc0 = __builtin_amdgcn_wmma_f32_16x16x4_f32(false, a, false, b0,
                                                     (short)0, acc0, false, false);
        acc1 = __builtin_amdgcn_wmma_f32_16x16x4_f32(false, a, false, b1v,
                                                     (short)0, acc1, false, false);
    }

    // epilogue: bias + relu, store H, per-row score + segment max
    const float bias0 = b1[l16], bias1 = b1[16 + l16];
    const float wc0 = wc[l16],   wc1 = wc[16 + l16];
    const float bcv = bc[0];
    const int rAdd = (lane < 16) ? 0 : 8;   // C/D layout: halves hold rows j and j+8

    for (int j = 0; j < 8; ++j) {
        float h0 = fmaxf(acc0[j] + bias0, 0.0f);
        float h1 = fmaxf(acc1[j] + bias1, 0.0f);
        long row = rowBase + j + rAdd;
        H[row * 32 + l16]      = h0;
        H[row * 32 + 16 + l16] = h1;
        float p = h0 * wc0 + h1 * wc1;
        p += __shfl_xor(p, 1, 32);
        p += __shfl_xor(p, 2, 32);
        p += __shfl_xor(p, 4, 32);
        p += __shfl_xor(p, 8, 32);          // full reduction within each 16-lane half
        if (l16 == 0) {
            float sv = p + bcv;
            S[row] = sv;
            atomicMax(&segMax[batch[row]], encf(sv));
        }
    }
}

// ---------------- K2: e = exp(s - max); segment denom ----------------
__global__ void k_soft(const int* __restrict__ batch, const unsigned* __restrict__ segMax,
                       float* __restrict__ S, float* __restrict__ segDen) {
    long i = (long)blockIdx.x * 256 + threadIdx.x;
    if (i >= N_ROWS) return;
    int seg = batch[i];
    float e = __expf(S[i] - decf(segMax[seg]));
    S[i] = e;
    atomicAdd(&segDen[seg], e);
}

// ---------------- K3: P[seg,:] += H[row,:] * (e/den) ----------------
__global__ void k_pool(const int* __restrict__ batch, const float* __restrict__ S,
                       const float* __restrict__ segDen, const float* __restrict__ H,
                       float* __restrict__ P) {
    long idx = (long)blockIdx.x * 256 + threadIdx.x;   // N_ROWS*32 threads
    long row = idx >> 5;
    int  c   = (int)(idx & 31);
    int seg = batch[row];
    float w = S[row] / segDen[seg];
    atomicAdd(&P[(long)seg * 32 + c], H[idx] * w);
}

// ---------------- K4: per-bag attention over 16 cell types ----------------
__global__ __launch_bounds__(256) void k_final(
        const float* __restrict__ P, const float* __restrict__ wct,
        const float* __restrict__ bct, const float* __restrict__ Wout,
        const float* __restrict__ bout, float* __restrict__ out) {
    int bag  = (blockIdx.x * 256 + threadIdx.x) >> 5;  // one wave per bag
    int lane = threadIdx.x & 31;
    if (bag >= NBAGS) return;
    const float* Pb = P + (long)bag * NCT * 32;

    float pv[NCT], t[NCT];
    const float wl = wct[lane];
    for (int ct = 0; ct < NCT; ++ct) {
        float v = Pb[ct * 32 + lane];
        pv[ct] = v;
        float s = v * wl;
        s += __shfl_xor(s, 1, 32);  s += __shfl_xor(s, 2, 32);
        s += __shfl_xor(s, 4, 32);  s += __shfl_xor(s, 8, 32);
        s += __shfl_xor(s, 16, 32);
        t[ct] = s + bct[0];
    }
    float m = t[0];
    for (int ct = 1; ct < NCT; ++ct) m = fmaxf(m, t[ct]);
    float den = 0.0f;
    for (int ct = 0; ct < NCT; ++ct) { t[ct] = __expf(t[ct] - m); den += t[ct]; }
    float inv = 1.0f / den;
    float xb = 0.0f;
    for (int ct = 0; ct < NCT; ++ct) xb += pv[ct] * t[ct];
    xb *= inv;
    float o = xb * Wout[lane];
    o += __shfl_xor(o, 1, 32);  o += __shfl_xor(o, 2, 32);
    o += __shfl_xor(o, 4, 32);  o += __shfl_xor(o, 8, 32);
    o += __shfl_xor(o, 16, 32);
    if (lane == 0) out[bag] = o + bout[0];
}

extern "C" void kernel_launch(void* const* d_in, const int* in_sizes, int n_in,
                              void* d_out, int out_size, void* d_ws, size_t ws_size,
                              hipStream_t stream) {
    (void)in_sizes; (void)n_in; (void)out_size; (void)ws_size;
    const float* X    = (const float*)d_in[0];
    const float* W1   = (const float*)d_in[1];
    const float* b1   = (const float*)d_in[2];
    const float* wc   = (const float*)d_in[3];
    const float* bc   = (const float*)d_in[4];
    const float* wct  = (const float*)d_in[5];
    const float* bct  = (const float*)d_in[6];
    const float* Wout = (const float*)d_in[7];
    const float* bout = (const float*)d_in[8];
    const int*   batch = (const int*)d_in[9];
    float* out = (float*)d_out;

    // workspace layout (floats): H[2M*32] | S[2M] | segMax[16384] | segDen[16384] | P[16384*32]
    float* ws = (float*)d_ws;
    float*    H      = ws;
    float*    S      = ws + (size_t)N_ROWS * N_HIDD;
    unsigned* segMax = (unsigned*)(S + N_ROWS);
    float*    segDen = (float*)(segMax + CT_SZ);
    float*    P      = segDen + CT_SZ;

    k_init <<< (CT_SZ * N_HIDD + 255) / 256, 256, 0, stream >>>(segMax, segDen, P);
    k_gemm <<< N_ROWS / 64, 128, 0, stream >>>(X, W1, b1, wc, bc, batch, H, S, segMax);
    k_soft <<< (N_ROWS + 255) / 256, 256, 0, stream >>>(batch, segMax, S, segDen);
    k_pool <<< N_ROWS / 8, 256, 0, stream >>>(batch, S, segDen, H, P);
    k_final<<< NBAGS / 8, 256, 0, stream >>>(P, wct, bct, Wout, bout, out);
}